// PerformanceModel_39840116638520
// MI455X (gfx1250) — compile-verified
//
#include <hip/hip_runtime.h>
#include <hip/hip_bf16.h>
#include <stdint.h>

#define NBINS_MAX 1000
#define BLOCK 256

// Native clang vector types (required by __builtin_nontemporal_*)
typedef int   vi4 __attribute__((ext_vector_type(4)));
typedef float vf4 __attribute__((ext_vector_type(4)));

// Async global->LDS copy of one 16-byte vector (CDNA5 async-tensor path).
// GVS mode: the global address is the 64-bit SGPR pair saddr plus the 32-bit byte offset in vaddr;
// the 16 bytes land at the LDS byte address held in vdst.
__device__ __forceinline__ void async_copy_b128(uint32_t lds_byte_addr,
                                                uint32_t glb_byte_off,
                                                const void* base) {
    asm volatile("global_load_async_to_lds_b128 %0, %1, %2"
                 :
                 : "v"(lds_byte_addr), "v"(glb_byte_off), "s"(base)
                 : "memory");
}

__device__ __forceinline__ void wait_asynccnt0() {
    asm volatile("s_wait_asynccnt 0" ::: "memory");
}

__global__ __launch_bounds__(BLOCK) void perf_model_kernel(
    const float* __restrict__ centers,   // [nbins]
    const int*   __restrict__ idx,       // [n, 3] int32
    const float* __restrict__ lbp,       // [3]
    const float* __restrict__ ubp,       // [3]
    float* __restrict__ out,             // [n]
    int n, int nbins) {

    __shared__ float s_cent[NBINS_MAX];          // raw bin centers
    __shared__ float s_tab[3 * NBINS_MAX];       // sigmoid table, bound-major

    const int tid = threadIdx.x;

    // ---- Stage 1: async cooperative fill of bin centers into LDS ----
    {
        const uint32_t lds_base = (uint32_t)(uintptr_t)(&s_cent[0]);
        const int nvec = (nbins * 4) / 16;       // # of full 16B vectors
        for (int i = tid; i < nvec; i += BLOCK) {
            const uint32_t off = (uint32_t)(i * 16);
            async_copy_b128(lds_base + off, off, (const void*)centers);
        }
        // scalar remainder (nbins not multiple of 4) via normal path
        for (int j = (nbins & ~3) + tid; j < nbins; j += BLOCK)
            s_cent[j] = centers[j];
        wait_asynccnt0();                        // each wave drains its own async ops
    }
    __syncthreads();

    // ---- Stage 2: build the 3 x nbins sigmoid table in LDS ----
    // lb/ub are 3 floats each; swap so lo<=hi, like the reference.
    float ubv[3], inv[3];
#pragma unroll
    for (int k = 0; k < 3; ++k) {
        const float a = lbp[k], b = ubp[k];
        const float lo = fminf(a, b), hi = fmaxf(a, b);
        ubv[k] = hi;
        inv[k] = 1.0f / (hi - lo + 1e-4f);
    }
    const int ntab = 3 * nbins;
    for (int e = tid; e < ntab; e += BLOCK) {
        const int k = e / nbins;
        const int j = e - k * nbins;
        const float c = s_cent[j];
        const float g = logf(c / (1.0f - c));            // logit
        const float x = (ubv[k] - g) * inv[k];
        s_tab[e] = 1.0f / (1.0f + expf(-x));             // sigmoid
    }
    __syncthreads();

    const float* __restrict__ T0 = s_tab;
    const float* __restrict__ T1 = s_tab + nbins;
    const float* __restrict__ T2 = s_tab + 2 * nbins;

    // ---- Stage 3: stream indices (NT), gather from LDS, 2 muls/row, store (NT) ----
    const int     n4    = n >> 2;                         // groups of 4 rows
    const long    gsz   = (long)gridDim.x * BLOCK;
    const vi4*    idx4  = (const vi4*)idx;
    vf4*          out4  = (vf4*)out;

    for (long t = (long)blockIdx.x * BLOCK + tid; t < n4; t += gsz) {
        // rows 4t..4t+3: 12 ints = 3 x b128, streamed once -> non-temporal
        const vi4 a = __builtin_nontemporal_load(&idx4[3 * t + 0]);
        const vi4 b = __builtin_nontemporal_load(&idx4[3 * t + 1]);
        const vi4 c = __builtin_nontemporal_load(&idx4[3 * t + 2]);
        vf4 r;
        r.x = T0[a.x] * T1[a.y] * T2[a.z];
        r.y = T0[a.w] * T1[b.x] * T2[b.y];
        r.z = T0[b.z] * T1[b.w] * T2[c.x];
        r.w = T0[c.y] * T1[c.z] * T2[c.w];
        __builtin_nontemporal_store(r, &out4[t]);
    }
    // scalar tail (n not multiple of 4)
    for (long r = (long)(n4 << 2) + (long)blockIdx.x * BLOCK + tid; r < n; r += gsz) {
        out[r] = T0[idx[3 * r]] * T1[idx[3 * r + 1]] * T2[idx[3 * r + 2]];
    }
}

extern "C" void kernel_launch(void* const* d_in, const int* in_sizes, int n_in,
                              void* d_out, int out_size, void* d_ws, size_t ws_size,
                              hipStream_t stream) {
    (void)n_in; (void)d_ws; (void)ws_size;
    const float* centers = (const float*)d_in[0];   // bin_centers [nbins] f32
    const int*   idx     = (const int*)d_in[1];     // obs idx [n,3] int32
    const float* lb      = (const float*)d_in[2];   // [3] f32
    const float* ub      = (const float*)d_in[3];   // [3] f32
    float*       out     = (float*)d_out;           // [n] f32

    const int nbins = in_sizes[0];
    const int n     = out_size;                     // n_diffs

    int n4 = n >> 2;
    int blocks = (n4 + BLOCK - 1) / BLOCK;          // 8192 for n = 8,388,608
    if (blocks < 1) blocks = 1;

    perf_model_kernel<<<blocks, BLOCK, 0, stream>>>(centers, idx, lb, ub, out, n, nbins);
}